// DySimGCF_18202071400771
// MI455X (gfx1250) — compile-verified
//
#include <hip/hip_runtime.h>
#include <stdint.h>

// =====================================================================
// DySimGCF edge-softmax message passing, tuned for MI455X (gfx1250).
//   incoming = segment_softmax(attr, to);  outgoing = segment_softmax(attr, from)
//   out[to] += sqrt(incoming*outgoing) * x[from]
//
// Memory-bound scatter/gather: x (25.6MB) + out (25.6MB) stay resident in
// the 192MB L2; edge streams get NT hints; scatter uses hardware f32
// atomics resolved at L2. Edge-index tiles are DMA'd into LDS via the
// CDNA5 Tensor Data Mover (TENSORcnt), edge attrs via the async
// global->LDS path (ASYNCcnt).
// =====================================================================

#define THREADS 256
#define FEAT    64
#define FG      16   // FEAT / 4 float4 groups

typedef unsigned int u32x4 __attribute__((ext_vector_type(4)));
typedef int          i32x4 __attribute__((ext_vector_type(4)));
typedef int          i32x8 __attribute__((ext_vector_type(8)));

// ---- order-preserving float<->uint encode for atomicMax on floats ----
__device__ __forceinline__ uint32_t f32_to_ordered(float f) {
  uint32_t u = __float_as_uint(f);
  return (u & 0x80000000u) ? ~u : (u | 0x80000000u);
}
__device__ __forceinline__ float ordered_to_f32(uint32_t k) {
  uint32_t u = (k & 0x80000000u) ? (k & 0x7fffffffu) : ~k;
  return __uint_as_float(u);
}

// ---- CDNA5 async global -> LDS copy (tracked by ASYNCcnt) ----
__device__ __forceinline__ void async_g2lds_b32(void* lds_dst, const void* gsrc) {
  asm volatile("global_load_async_to_lds_b32 %0, %1, off"
               :: "v"((uint32_t)(uintptr_t)lds_dst), "v"(gsrc)
               : "memory");
}
__device__ __forceinline__ void wait_async0() {
  asm volatile("s_wait_asynccnt 0" ::: "memory");
}

// ---- CDNA5 Tensor Data Mover: 2D tile (4-byte elems) global -> LDS ----
// Descriptor bitfields per CDNA5 ISA ch.10.8 (D# groups 0/1; groups 2/3
// zero => dims 2+ unused).  tensor_d0 clips the tail tile (OOB reads = 0).
__device__ __forceinline__ void tdm_load_2d(uint32_t lds_byte_off,
                                            const void* gaddr,
                                            uint32_t tensor_d0,  // elems to tensor end
                                            uint32_t tensor_d1,  // rows
                                            uint32_t tile_d0,    // tile width (elems)
                                            uint32_t tile_d1,    // tile rows
                                            uint64_t row_stride) // elems
{
  uint64_t ga = (uint64_t)(uintptr_t)gaddr;
  u32x4 g0;
  g0[0] = 0x1u;                                  // count=1 (valid), no gather
  g0[1] = lds_byte_off;                          // lds_addr
  g0[2] = (uint32_t)(ga & 0xffffffffu);          // global_addr[31:0]
  g0[3] = (uint32_t)((ga >> 32) & 0x01ffffffu)   // global_addr[56:32]
        | 0x80000000u;                           // type=2 in bits[127:126]
  i32x8 g1;
  g1[0] = (int)(2u << 16);                       // data_size=4B; mask/flags=0
  g1[1] = (int)((tensor_d0 & 0xffffu) << 16);    // tensor_dim0[15:0]
  g1[2] = (int)((tensor_d0 >> 16) | ((tensor_d1 & 0xffffu) << 16));
  g1[3] = (int)((tensor_d1 >> 16) | (tile_d0 << 16));   // tile_dim0
  g1[4] = (int)(tile_d1 & 0xffffu);              // tile_dim1 (tile_dim2=0)
  g1[5] = (int)(uint32_t)(row_stride & 0xffffffffu);    // dim0_stride[31:0]
  g1[6] = (int)(uint32_t)((row_stride >> 32) & 0xffffu);
  g1[7] = 0;
  i32x4 z4 = {0, 0, 0, 0};
#if __clang_major__ >= 23
  i32x8 z8 = {0, 0, 0, 0, 0, 0, 0, 0};
  __builtin_amdgcn_tensor_load_to_lds(g0, g1, z4, z4, z8, 0);
#else
  __builtin_amdgcn_tensor_load_to_lds(g0, g1, z4, z4, 0);
#endif
}

// ---------------------------------------------------------------------
// One block = one 256-edge tile. edge_index tile (both rows) arrives via
// one TDM descriptor, attrs via per-lane async loads; then 16 lanes
// cooperate per edge (float4 per lane = 64 features).
__global__ void __launch_bounds__(THREADS)
k_scatter(const float* __restrict__ x,
          const int* __restrict__ src, const int* __restrict__ dst,
          const float* __restrict__ attr,
          const uint32_t* __restrict__ max_in, const uint32_t* __restrict__ max_out,
          const float* __restrict__ sum_in, const float* __restrict__ sum_out,
          float* __restrict__ out, int E) {
  __shared__ int   s_idx[2 * THREADS];   // [0..255]=from row, [256..511]=to row
  __shared__ float s_attr[THREADS];

  const int tid  = threadIdx.x;
  const int base = blockIdx.x * THREADS;

  {  // attrs: CDNA5 async global->LDS (clamped tail)
    int e  = base + tid;
    int es = (e < E) ? e : (E - 1);
    async_g2lds_b32(&s_attr[tid], attr + es);
  }
  if (tid < 32) {  // edge_index tile: single TDM DMA, wave 0 issues
    tdm_load_2d((uint32_t)(uintptr_t)&s_idx[0], src + base,
                (uint32_t)(E - base), 2u, (uint32_t)THREADS, 2u, (uint64_t)E);
    __builtin_amdgcn_s_wait_tensorcnt(0);
  }
  wait_async0();
  __syncthreads();

  const int g   = tid & 15;   // float4 feature group 0..15
  const int sub = tid >> 4;   // 16 concurrent edges per pass
  const float4* __restrict__ x4 = (const float4*)x;

  #pragma unroll 1
  for (int i = 0; i < 16; ++i) {
    int le = i * 16 + sub;
    int e  = base + le;
    if (e < E) {
      // prefetch next pass's gathered x row (keeps WGP$/L2 warm)
      int le2 = le + 16;
      if (le2 < THREADS && (base + le2) < E)
        __builtin_prefetch(&x4[(size_t)s_idx[le2] * FG + g], 0, 3);

      int   f = s_idx[le];
      int   t = s_idx[THREADS + le];
      float a = s_attr[le];

      float ei   = expf(a - ordered_to_f32(max_in[t]));
      float eo   = expf(a - ordered_to_f32(max_out[f]));
      float norm = sqrtf((ei * eo) / (sum_in[t] * sum_out[f]));

      float4 v = x4[(size_t)f * FG + g];
      float* o = out + (size_t)t * FEAT + (g << 2);
      unsafeAtomicAdd(o + 0, norm * v.x);
      unsafeAtomicAdd(o + 1, norm * v.y);
      unsafeAtomicAdd(o + 2, norm * v.z);
      unsafeAtomicAdd(o + 3, norm * v.w);
    }
  }
}

// ---------------------------------------------------------------------
__global__ void __launch_bounds__(THREADS)
k_init(uint32_t* __restrict__ max_in, uint32_t* __restrict__ max_out,
       float* __restrict__ sum_in, float* __restrict__ sum_out,
       float* __restrict__ out, int n_nodes, int out_elems) {
  int i = blockIdx.x * THREADS + threadIdx.x;
  if (i < n_nodes) {
    max_in[i]  = 0u;   // ordered-encoding identity (< any real float's key)
    max_out[i] = 0u;
    sum_in[i]  = 0.0f;
    sum_out[i] = 0.0f;
  }
  if (i < out_elems) out[i] = 0.0f;
}

// ---------------------------------------------------------------------
__global__ void __launch_bounds__(THREADS)
k_edge_max(const int* __restrict__ src, const int* __restrict__ dst,
           const float* __restrict__ attr,
           uint32_t* __restrict__ max_in, uint32_t* __restrict__ max_out, int E) {
  int e = blockIdx.x * THREADS + threadIdx.x;
  if (e >= E) return;
  int f = __builtin_nontemporal_load(src + e);
  int t = __builtin_nontemporal_load(dst + e);
  uint32_t k = f32_to_ordered(__builtin_nontemporal_load(attr + e));
  atomicMax(&max_in[t], k);
  atomicMax(&max_out[f], k);
}

// ---------------------------------------------------------------------
__global__ void __launch_bounds__(THREADS)
k_edge_sum(const int* __restrict__ src, const int* __restrict__ dst,
           const float* __restrict__ attr,
           const uint32_t* __restrict__ max_in, const uint32_t* __restrict__ max_out,
           float* __restrict__ sum_in, float* __restrict__ sum_out, int E) {
  int e = blockIdx.x * THREADS + threadIdx.x;
  if (e >= E) return;
  int   f = __builtin_nontemporal_load(src + e);
  int   t = __builtin_nontemporal_load(dst + e);
  float a = __builtin_nontemporal_load(attr + e);
  unsafeAtomicAdd(&sum_in[t],  expf(a - ordered_to_f32(max_in[t])));
  unsafeAtomicAdd(&sum_out[f], expf(a - ordered_to_f32(max_out[f])));
}

// ---------------------------------------------------------------------
extern "C" void kernel_launch(void* const* d_in, const int* in_sizes, int n_in,
                              void* d_out, int out_size, void* d_ws, size_t ws_size,
                              hipStream_t stream) {
  (void)n_in; (void)ws_size;
  const float* x   = (const float*)d_in[0];
  const int*   ei  = (const int*)d_in[1];
  const float* ea  = (const float*)d_in[2];

  const int E = in_sizes[2];          // edge_attrs count
  const int N = out_size / FEAT;      // nodes
  const int* src = ei;                // edge_index[0]
  const int* dst = ei + E;            // edge_index[1]
  float* out = (float*)d_out;

  // scratch layout in d_ws: [max_in u32 | max_out u32 | sum_in f32 | sum_out f32]
  uint32_t* max_in  = (uint32_t*)d_ws;
  uint32_t* max_out = max_in + N;
  float*    sum_in  = (float*)(max_out + N);
  float*    sum_out = sum_in + N;

  const int init_blocks = (out_size + THREADS - 1) / THREADS;
  const int edge_blocks = (E + THREADS - 1) / THREADS;

  k_init<<<init_blocks, THREADS, 0, stream>>>(max_in, max_out, sum_in, sum_out,
                                              out, N, out_size);
  k_edge_max<<<edge_blocks, THREADS, 0, stream>>>(src, dst, ea, max_in, max_out, E);
  k_edge_sum<<<edge_blocks, THREADS, 0, stream>>>(src, dst, ea, max_in, max_out,
                                                  sum_in, sum_out, E);
  k_scatter<<<edge_blocks, THREADS, 0, stream>>>(x, src, dst, ea, max_in, max_out,
                                                 sum_in, sum_out, out, E);
}